// SingleScalarGCN_51384988729601
// MI455X (gfx1250) — compile-verified
//
#include <hip/hip_runtime.h>
#include <math.h>

typedef __attribute__((ext_vector_type(2))) float v2f;
typedef __attribute__((ext_vector_type(8))) float v8f;

#define N_NODES 50000
#define NFEAT   256
#define NHID    128
#define NOUT    64
#define LDSTRIDE 258   // 256 + 2-float pad -> conflict-free ds_load A-frags

// ---------------------------------------------------------------------------
// Edge-parallel SpMM accumulate: acc[rows[e]][t] += s*vals[e]*h[cols[e]][t]
// thread t = feature index (0..127): fully coalesced 512B gather per edge.
// Scatter uses the hardware no-return fp32 global atomic (STOREcnt-tracked)
// issued directly via inline asm so it cannot CAS-expand.
// ---------------------------------------------------------------------------
__global__ __launch_bounds__(128) void gcn_spmm(const int* __restrict__ rows,
                                                const int* __restrict__ cols,
                                                const float* __restrict__ vals,
                                                const float* __restrict__ scalar,
                                                const float* __restrict__ h,
                                                float* __restrict__ acc,
                                                int nedges)
{
    const int t = threadIdx.x;                    // feature
    const float s = scalar ? scalar[0] : 1.0f;
    int e0 = blockIdx.x * 32;
    int e1 = e0 + 32;
    if (e1 > nedges) e1 = nedges;
    for (int e = e0; e < e1; ++e) {
        const int r = rows[e];
        const int cidx = cols[e];
        const float v = vals[e] * s;
        const float contrib = v * h[(size_t)cidx * NHID + t];
        float* p = &acc[(size_t)r * NHID + t];
        asm volatile("global_atomic_add_f32 %0, %1, off"
                     :: "v"(p), "v"(contrib)
                     : "memory");
    }
}

// ---------------------------------------------------------------------------
// GEMM1: H[i][j] = sum_k x[i][k] * W1[j][k] + b1[j]   (i<50000, j<128, k<256)
// The 16x256 x-tile is staged once per workgroup into LDS with CDNA5 async
// global->LDS copies (ASYNCcnt), then all 8 waves feed their WMMA A-frags
// from LDS; B-frags (W1, L2-resident) come straight from global.
// A-frag (16x4 f32): lane L -> row M=L&15, K-pair koff=(L>>4)*2 (v0=K, v1=K+1)
// B-frag (4x16 f32): lane L -> col N=L&15, same K-pair layout.
// C/D   (16x16 f32): VGPR r on lane L -> M = r + 8*(L>>4), N = L&15.
// ---------------------------------------------------------------------------
__global__ __launch_bounds__(256) void gcn_gemm1(const float* __restrict__ x,
                                                 const float* __restrict__ W1,
                                                 const float* __restrict__ b1,
                                                 float* __restrict__ H)
{
    __shared__ float tile[16 * LDSTRIDE];

    const int tid  = threadIdx.x;
    const int lane = tid & 31;
    const int wave = tid >> 5;                    // j-tile 0..7
    const int row0 = blockIdx.x << 4;             // 16 node rows per block

    // ---- async stage: 16 rows x 256 floats, 64B (4 x b128) per thread ----
    {
        const int srow = tid >> 4;                // 0..15
        const int sseg = tid & 15;                // 0..15 (16 floats each)
        const float* gp = x + (size_t)(row0 + srow) * NFEAT + sseg * 16;
        const float* lp = &tile[srow * LDSTRIDE + sseg * 16];
        unsigned ldsb = (unsigned)(size_t)lp;     // low 32 bits = LDS byte addr
#pragma unroll
        for (int q = 0; q < 4; ++q) {
            asm volatile("global_load_async_to_lds_b128 %0, %1, off"
                         :: "v"(ldsb + 16u * q), "v"(gp + 4 * q)
                         : "memory");
        }
        asm volatile("s_wait_asynccnt 0x0" ::: "memory");
    }
    __syncthreads();

    const int m    = lane & 15;
    const int koff = (lane >> 4) << 1;            // 0 or 2
    const int j    = (wave << 4) + (lane & 15);   // output column (B/C/D lane)

    const float* arow = &tile[m * LDSTRIDE];      // LDS, bank-padded
    const float* wrow = W1 + (size_t)j * NFEAT;   // W1 is [NHID, NFEAT] row-major

    v8f c;
    const float bj = b1[j];
#pragma unroll
    for (int r = 0; r < 8; ++r) c[r] = bj;

#pragma unroll 8
    for (int k = 0; k < NFEAT; k += 4) {
        v2f a = *(const v2f*)(arow + k + koff);   // ds_load (2addr-merged)
        v2f b = *(const v2f*)(wrow + k + koff);   // global_load_b64
        c = __builtin_amdgcn_wmma_f32_16x16x4_f32(false, a, false, b,
                                                  (short)0, c, false, false);
    }

    const int mh = (lane >> 4) << 3;              // 0 or 8
    float* out = H + (size_t)(row0 + mh) * NHID + j;
#pragma unroll
    for (int r = 0; r < 8; ++r) out[(size_t)r * NHID] = c[r];
}

// ---------------------------------------------------------------------------
// GEMM2: out[i][j] = sum_k h[i][k] * Wout[j][k] + bout[j]  (j<64, k<128)
// 4 waves/block cover the 64 output columns; h rows are L2-hot (just written
// by the ELU pass), so plain global fragment loads are fine here.
// ---------------------------------------------------------------------------
__global__ __launch_bounds__(128) void gcn_gemm2(const float* __restrict__ h,
                                                 const float* __restrict__ Wout,
                                                 const float* __restrict__ bout,
                                                 float* __restrict__ out)
{
    const int lane = threadIdx.x & 31;
    const int wave = threadIdx.x >> 5;            // j-tile 0..3
    const int row0 = blockIdx.x << 4;
    const int m    = lane & 15;
    const int koff = (lane >> 4) << 1;
    const int j    = (wave << 4) + (lane & 15);

    const float* hrow = h    + (size_t)(row0 + m) * NHID;
    const float* wrow = Wout + (size_t)j * NHID;  // Wout is [NOUT, NHID] row-major

    v8f c;
    const float bj = bout[j];
#pragma unroll
    for (int r = 0; r < 8; ++r) c[r] = bj;

#pragma unroll 8
    for (int k = 0; k < NHID; k += 4) {
        v2f a = *(const v2f*)(hrow + k + koff);
        v2f b = *(const v2f*)(wrow + k + koff);
        c = __builtin_amdgcn_wmma_f32_16x16x4_f32(false, a, false, b,
                                                  (short)0, c, false, false);
    }

    const int mh = (lane >> 4) << 3;
    float* op = out + (size_t)(row0 + mh) * NOUT + j;
#pragma unroll
    for (int r = 0; r < 8; ++r) op[(size_t)r * NOUT] = c[r];
}

__global__ void gcn_zero(float* __restrict__ a, int n)
{
    int i = blockIdx.x * blockDim.x + threadIdx.x;
    const int stride = gridDim.x * blockDim.x;
    for (; i < n; i += stride) a[i] = 0.0f;
}

__global__ void gcn_elu(float* __restrict__ a, int n)
{
    int i = blockIdx.x * blockDim.x + threadIdx.x;
    const int stride = gridDim.x * blockDim.x;
    for (; i < n; i += stride) {
        const float v = a[i];
        a[i] = (v > 0.0f) ? v : (expf(v) - 1.0f);
    }
}

extern "C" void kernel_launch(void* const* d_in, const int* in_sizes, int n_in,
                              void* d_out, int out_size, void* d_ws, size_t ws_size,
                              hipStream_t stream)
{
    const float* x    = (const float*)d_in[0];
    const int*   rows = (const int*)  d_in[1];
    const int*   cols = (const int*)  d_in[2];
    const float* vals = (const float*)d_in[3];
    const float* W1   = (const float*)d_in[4];
    const float* b1   = (const float*)d_in[5];
    const float* scal = (const float*)d_in[6];
    const float* Wout = (const float*)d_in[7];
    const float* bout = (const float*)d_in[8];
    float* out = (float*)d_out;

    const int E  = in_sizes[1];
    const int HN = N_NODES * NHID;

    float* bufA = (float*)d_ws;
    float* bufB = bufA + (size_t)N_NODES * NHID;

    const int zgrid = 1024;
    const int sgrid = (E + 31) / 32;
    const int ggrid = N_NODES / 16;               // 50000 = 3125 * 16 exactly

    // layer 0 linear: bufA = x @ W1^T + b1
    gcn_gemm1<<<ggrid, 256, 0, stream>>>(x, W1, b1, bufA);

    // layer 0 aggregate + ELU: bufB = elu(A @ bufA)
    gcn_zero<<<zgrid, 256, 0, stream>>>(bufB, HN);
    gcn_spmm<<<sgrid, 128, 0, stream>>>(rows, cols, vals, nullptr, bufA, bufB, E);
    gcn_elu<<<zgrid, 256, 0, stream>>>(bufB, HN);

    // layer 1: bufA = elu(A @ (scalar * bufB))
    gcn_zero<<<zgrid, 256, 0, stream>>>(bufA, HN);
    gcn_spmm<<<sgrid, 128, 0, stream>>>(rows, cols, vals, scal, bufB, bufA, E);
    gcn_elu<<<zgrid, 256, 0, stream>>>(bufA, HN);

    // layer 2: bufB = elu(A @ (scalar * bufA))
    gcn_zero<<<zgrid, 256, 0, stream>>>(bufB, HN);
    gcn_spmm<<<sgrid, 128, 0, stream>>>(rows, cols, vals, scal, bufA, bufB, E);
    gcn_elu<<<zgrid, 256, 0, stream>>>(bufB, HN);

    // output projection: out = bufB @ Wout^T + bout
    gcn_gemm2<<<ggrid, 128, 0, stream>>>(bufB, Wout, bout, out);
}